// BDHGPURefTorch_53807350284324
// MI455X (gfx1250) — compile-verified
//
#include <hip/hip_runtime.h>

// Problem constants (from reference)
#define Nn 16384
#define Dd 256
#define Vv 32000
#define Tt 128
#define DECAY 0.97f

typedef __attribute__((ext_vector_type(2))) float v2f;
typedef __attribute__((ext_vector_type(8))) float v8f;

// ---------------------------------------------------------------------------
// K0: gather Vprev[t][d] = token_emb[tokens[t]][d]
//     and VwT[d][t] = 0.97^(128-t) * Vprev[t][d]  (A-matrix for rho GEMM)
// ---------------------------------------------------------------------------
__global__ void k0_gather(const float* __restrict__ token_emb,
                          const int* __restrict__ tokens,
                          float* __restrict__ Vprev, float* __restrict__ VwT) {
    int t = blockIdx.x;      // 0..127
    int d = threadIdx.x;     // 0..255
    int tok = tokens[t];
    float v = token_emb[tok * Dd + d];
    Vprev[t * Dd + d] = v;
    float w = powf(DECAY, (float)(Tt - t));   // rho_f weight for step t
    VwT[d * Tt + t] = w * v;
}

// ---------------------------------------------------------------------------
// K1: R = relu(Dx @ Vprev^T)  -> X[n][t]   (16384x128), WMMA f32 16x16x4
//     one wave per 16x16 output tile; K-loop over D=256 in steps of 4.
//   A[m][k] = Dx[nTile*16+m][k] : lane (m=lane%16), regs k = 4kk+2*(lane/16)+r
//   B[k][c] = Vprev[tTile*16+c][k] (B = Vprev^T), mirrored layout
// ---------------------------------------------------------------------------
__global__ void k1_gemm_relu(const float* __restrict__ Dx,
                             const float* __restrict__ Vprev,
                             float* __restrict__ X) {
    int wave   = (blockIdx.x * blockDim.x + threadIdx.x) >> 5;
    int lane   = threadIdx.x & 31;
    int half   = lane >> 4;
    int within = lane & 15;
    int nTile  = wave >> 3;     // 0..1023
    int tTile  = wave & 7;      // 0..7

    const float* arow = Dx    + (size_t)(nTile * 16 + within) * Dd;
    const float* brow = Vprev + (size_t)(tTile * 16 + within) * Dd;

    v8f c = {};
#pragma unroll 4
    for (int kk = 0; kk < Dd / 4; ++kk) {
        int k0 = kk * 4 + half * 2;
        v2f a = *(const v2f*)(arow + k0);
        v2f b = *(const v2f*)(brow + k0);
        c = __builtin_amdgcn_wmma_f32_16x16x4_f32(false, a, false, b,
                                                  (short)0, c, false, false);
    }
    int tcol = tTile * 16 + within;
#pragma unroll
    for (int r = 0; r < 8; ++r) {
        int nrow = nTile * 16 + r + 8 * half;
        float val = c[r];
        X[(size_t)nrow * Tt + tcol] = val > 0.f ? val : 0.f;   // fused relu
    }
}

// ---------------------------------------------------------------------------
// K2: inclusive cumsum over t per row (x_t recurrence); x_f -> d_out[0:N)
// ---------------------------------------------------------------------------
__global__ void k2_cumsum(float* __restrict__ X, float* __restrict__ xf_out) {
    int n = blockIdx.x * blockDim.x + threadIdx.x;
    float* row = X + (size_t)n * Tt;
    float acc = 0.f;
    for (int t = 0; t < Tt; ++t) { acc += row[t]; row[t] = acc; }
    xf_out[n] = acc;
}

// ---------------------------------------------------------------------------
// K3: rho_f[d][n] = sum_t VwT[d][t] * X[n][t]   (256x16384), WMMA f32 16x16x4
// ---------------------------------------------------------------------------
__global__ void k3_rho(const float* __restrict__ VwT,
                       const float* __restrict__ X,
                       float* __restrict__ rho) {
    int wave   = (blockIdx.x * blockDim.x + threadIdx.x) >> 5;
    int lane   = threadIdx.x & 31;
    int half   = lane >> 4;
    int within = lane & 15;
    int dTile  = wave >> 10;    // 0..15
    int nTile  = wave & 1023;   // 0..1023

    const float* arow = VwT + (size_t)(dTile * 16 + within) * Tt;
    const float* brow = X   + (size_t)(nTile * 16 + within) * Tt;

    v8f c = {};
#pragma unroll 4
    for (int kk = 0; kk < Tt / 4; ++kk) {
        int k0 = kk * 4 + half * 2;
        v2f a = *(const v2f*)(arow + k0);
        v2f b = *(const v2f*)(brow + k0);
        c = __builtin_amdgcn_wmma_f32_16x16x4_f32(false, a, false, b,
                                                  (short)0, c, false, false);
    }
    int ncol = nTile * 16 + within;
#pragma unroll
    for (int r = 0; r < 8; ++r) {
        int drow = dTile * 16 + r + 8 * half;
        rho[(size_t)drow * Nn + ncol] = c[r];
    }
}

// ---------------------------------------------------------------------------
// K4: g[s] = x_s . x_{T-1}   (block per s, reduce over N)
// ---------------------------------------------------------------------------
__global__ void k4_gram(const float* __restrict__ X, const float* __restrict__ xf,
                        float* __restrict__ g) {
    __shared__ float red[256];
    int s = blockIdx.x;
    float p = 0.f;
    for (int n = threadIdx.x; n < Nn; n += 256)
        p += X[(size_t)n * Tt + s] * xf[n];
    red[threadIdx.x] = p;
    __syncthreads();
    for (int off = 128; off > 0; off >>= 1) {
        if (threadIdx.x < off) red[threadIdx.x] += red[threadIdx.x + off];
        __syncthreads();
    }
    if (threadIdx.x == 0) g[s] = red[0];
}

// ---------------------------------------------------------------------------
// K5: a*[d] = sum_{s<=T-2} 0.97^(127-s) * g[s] * Vprev[s][d]; then row-LN
//     (torch-style: (z-mean)/(unbiased_std + 1e-6)) -> ln_a[256]
// ---------------------------------------------------------------------------
__global__ void k5_astar_ln(const float* __restrict__ g,
                            const float* __restrict__ Vprev,
                            float* __restrict__ ln_a) {
    __shared__ float gs[Tt];
    __shared__ float red[256];
    int d = threadIdx.x;
    if (d < Tt) gs[d] = g[d];
    __syncthreads();
    float acc = 0.f;
    float w = DECAY;                       // 0.97^(127-s), s = 126 .. 0
    for (int s = Tt - 2; s >= 0; --s) {
        acc += w * gs[s] * Vprev[s * Dd + d];
        w *= DECAY;
    }
    red[d] = acc; __syncthreads();
    for (int off = 128; off > 0; off >>= 1) {
        if (d < off) red[d] += red[d + off];
        __syncthreads();
    }
    float mean = red[0] / (float)Dd;
    __syncthreads();
    float dev = acc - mean;
    red[d] = dev * dev; __syncthreads();
    for (int off = 128; off > 0; off >>= 1) {
        if (d < off) red[d] += red[d + off];
        __syncthreads();
    }
    float stdv = sqrtf(red[0] / (float)(Dd - 1));
    ln_a[d] = dev / (stdv + 1e-6f);
}

// ---------------------------------------------------------------------------
// K6: y[n] = relu(Dy[n,:] . ln_a) * max(x_f[n], 0)  -> d_out[N:2N) and ws
//     one wave per row, lane-strided dot + shuffle reduce
// ---------------------------------------------------------------------------
__global__ void k6_y(const float* __restrict__ Dy, const float* __restrict__ ln_a,
                     const float* __restrict__ xf,
                     float* __restrict__ y_out, float* __restrict__ y_ws) {
    int n    = (blockIdx.x * blockDim.x + threadIdx.x) >> 5;  // row
    int lane = threadIdx.x & 31;
    const float* row = Dy + (size_t)n * Dd;
    float p = 0.f;
#pragma unroll
    for (int j = 0; j < Dd / 32; ++j)
        p += row[lane + 32 * j] * ln_a[lane + 32 * j];
    for (int off = 16; off > 0; off >>= 1)
        p += __shfl_down(p, off, 32);
    if (lane == 0) {
        float yc = p > 0.f ? p : 0.f;
        float xv = xf[n];
        xv = xv > 0.f ? xv : 0.f;
        float y = yc * xv;
        y_out[n] = y;
        y_ws[n]  = y;
    }
}

// ---------------------------------------------------------------------------
// K7: u[d] = E[d,:] . y  (block per d, reduce over N)
// ---------------------------------------------------------------------------
__global__ void k7_u(const float* __restrict__ E, const float* __restrict__ y,
                     float* __restrict__ u) {
    __shared__ float red[256];
    int d = blockIdx.x;
    float p = 0.f;
    for (int n = threadIdx.x; n < Nn; n += 256)
        p += E[(size_t)d * Nn + n] * y[n];
    red[threadIdx.x] = p; __syncthreads();
    for (int off = 128; off > 0; off >>= 1) {
        if (threadIdx.x < off) red[threadIdx.x] += red[threadIdx.x + off];
        __syncthreads();
    }
    if (threadIdx.x == 0) u[d] = red[0];
}

// ---------------------------------------------------------------------------
// K8: v* = LN(u) -> d_out[2N : 2N+D)
// ---------------------------------------------------------------------------
__global__ void k8_ln(const float* __restrict__ u, float* __restrict__ out) {
    __shared__ float red[256];
    int d = threadIdx.x;
    float z = u[d];
    red[d] = z; __syncthreads();
    for (int off = 128; off > 0; off >>= 1) {
        if (d < off) red[d] += red[d + off];
        __syncthreads();
    }
    float mean = red[0] / (float)Dd;
    __syncthreads();
    float dev = z - mean;
    red[d] = dev * dev; __syncthreads();
    for (int off = 128; off > 0; off >>= 1) {
        if (d < off) red[d] += red[d + off];
        __syncthreads();
    }
    float stdv = sqrtf(red[0] / (float)(Dd - 1));
    out[d] = dev / (stdv + 1e-6f);
}

// ---------------------------------------------------------------------------
// Workspace layout (bytes):  total ~8.72 MB
//   X     @ 0         : N*T*4   = 8,388,608
//   Vprev @ 8388608   : T*D*4   =   131,072
//   VwT   @ 8519680   : D*T*4   =   131,072
//   g     @ 8650752   : T*4     =       512
//   ln_a  @ 8651264   : D*4     =     1,024
//   y     @ 8652288   : N*4     =    65,536
//   u     @ 8717824   : D*4     =     1,024
// ---------------------------------------------------------------------------
extern "C" void kernel_launch(void* const* d_in, const int* in_sizes, int n_in,
                              void* d_out, int out_size, void* d_ws, size_t ws_size,
                              hipStream_t stream) {
    const float* E         = (const float*)d_in[0];   // (D, N)
    const float* Dx        = (const float*)d_in[1];   // (N, D)
    const float* Dy        = (const float*)d_in[2];   // (N, D)
    const float* token_emb = (const float*)d_in[3];   // (V, D)
    const int*   tokens    = (const int*)d_in[4];     // (T,)

    float* out   = (float*)d_out;
    float* xf    = out;                    // x_f          [N]
    float* yout  = out + Nn;               // ys[-1]       [N]
    float* vstar = out + 2 * Nn;           // vs[-1]       [D]
    float* rho   = out + 2 * Nn + Dd;      // rho_f        [D*N]

    char*  ws    = (char*)d_ws;
    float* X     = (float*)(ws + 0);
    float* Vprev = (float*)(ws + 8388608);
    float* VwT   = (float*)(ws + 8519680);
    float* g     = (float*)(ws + 8650752);
    float* ln_a  = (float*)(ws + 8651264);
    float* y_ws  = (float*)(ws + 8652288);
    float* u     = (float*)(ws + 8717824);

    // 1) embedding gather + decay-scaled A for rho GEMM
    k0_gather<<<Tt, Dd, 0, stream>>>(token_emb, tokens, Vprev, VwT);
    // 2) X = relu(Dx @ Vprev^T)   [WMMA]  (1024 x 8 tiles, 4 waves/block)
    k1_gemm_relu<<<2048, 128, 0, stream>>>(Dx, Vprev, X);
    // 3) inclusive cumsum over t -> x_t columns; x_f to d_out
    k2_cumsum<<<Nn / 256, 256, 0, stream>>>(X, xf);
    // 4) rho_f = VwT @ X^T        [WMMA]  (16 x 1024 tiles)
    k3_rho<<<4096, 128, 0, stream>>>(VwT, X, rho);
    // 5) g[s] = x_s . x_{T-1}
    k4_gram<<<Tt, 256, 0, stream>>>(X, xf, g);
    // 6) a* and layernorm
    k5_astar_ln<<<1, Dd, 0, stream>>>(g, Vprev, ln_a);
    // 7) y = relu(Dy @ ln_a) * max(x_f, 0)
    k6_y<<<4096, 128, 0, stream>>>(Dy, ln_a, xf, yout, y_ws);
    // 8) u = E @ y
    k7_u<<<Dd, 256, 0, stream>>>(E, y_ws, u);
    // 9) v* = LN(u)
    k8_ln<<<1, Dd, 0, stream>>>(u, vstar);
}